// MeshLoss2D_5583457484974
// MI455X (gfx1250) — compile-verified
//
#include <hip/hip_runtime.h>

typedef __attribute__((ext_vector_type(2))) float v2f;
typedef __attribute__((ext_vector_type(8))) float v8f;

#define H0 48
#define W0 48
#define HR 95
#define WR 95
#define QMESH (HR * WR)   // 9025
#define PPC   8192
#define NB    2

// ---------------------------------------------------------------------------
// 1) Mesh refinement: vertices [n,3,48,48] -> point list mesh [n, 9025, 3]
// ---------------------------------------------------------------------------
__global__ void refine_kernel(const float* __restrict__ V, float* __restrict__ mesh) {
    int t = blockIdx.x * blockDim.x + threadIdx.x;
    const int total = NB * QMESH;
    if (t >= total) return;
    int n = t / QMESH;
    int q = t % QMESH;
    int h = q / WR, w = q % WR;
    int i = h >> 1, j = w >> 1;
    bool ho = (h & 1) != 0, wo = (w & 1) != 0;
    #pragma unroll
    for (int c = 0; c < 3; ++c) {
        const float* plane = V + ((size_t)(n * 3 + c) * H0) * W0;
        float v00 = plane[i * W0 + j];
        float val;
        if (!ho && !wo) {
            val = v00;
        } else if (!ho && wo) {
            val = 0.5f * (v00 + plane[i * W0 + j + 1]);
        } else if (ho && !wo) {
            val = 0.5f * (v00 + plane[(i + 1) * W0 + j]);
        } else {
            val = 0.25f * (v00 + plane[i * W0 + j + 1] +
                           plane[(i + 1) * W0 + j] + plane[(i + 1) * W0 + j + 1]);
        }
        mesh[((size_t)n * QMESH + q) * 3 + c] = val;
    }
}

// ---------------------------------------------------------------------------
// 2) Pack WMMA B panels: for each 16-column tile of Y, emit the per-lane v2f
//    operand for V_WMMA_F32_16X16X4_F32 (B is 4x16, K-striped):
//      lanes 0-15 : (K0,K1) = (-2*yx[N], -2*yy[N])      N = lane
//      lanes 16-31: (K2,K3) = (-2*yz[N], |y[N]|^2)      N = lane-16
//    Tail columns get |y|^2 = 1e30 so they never win the min.
//    Layout: panels[n][tile][lane] : v2f
// ---------------------------------------------------------------------------
__global__ void pack_kernel(const float* __restrict__ Y, int S, int stiles,
                            float* __restrict__ panels) {
    int t = blockIdx.x * blockDim.x + threadIdx.x;
    const int total = NB * stiles * 32;
    if (t >= total) return;
    int lane = t & 31;
    int tile = (t >> 5) % stiles;
    int n    = t / (stiles * 32);
    int nl = lane & 15;
    int hi = lane >> 4;

    int  col   = tile * 16 + nl;
    bool valid = col < S;
    int  colc  = valid ? col : (S - 1);
    const float* yp = Y + ((size_t)n * S + colc) * 3;
    float yx = yp[0], yy = yp[1], yz = yp[2];
    float y2 = valid ? (yx * yx + yy * yy + yz * yz) : 1.0e30f;

    float p0 = hi ? (-2.0f * yz) : (-2.0f * yx);
    float p1 = hi ? y2           : (-2.0f * yy);
    float* dst = panels + (((size_t)n * stiles + tile) * 32 + lane) * 2;
    dst[0] = p0;
    dst[1] = p1;
}

// ---------------------------------------------------------------------------
// 3) One-directional Chamfer: each wave owns 32 X rows (two 16-row A tiles)
//    and sweeps all prepacked Y panels; two WMMAs per b64 panel load.
//    D = -2 x.y + |y|^2 ; |x|^2 (row-constant) added after the min-reduce.
// ---------------------------------------------------------------------------
__global__ __launch_bounds__(128)
void nearest_kernel(const float* __restrict__ X, int R,
                    const float* __restrict__ panels, int stiles,
                    float* __restrict__ outMin, int batches, int tilesPerBatch) {
    const int lane = threadIdx.x & 31;
    const int wave = threadIdx.x >> 5;
    int gw = blockIdx.x * 4 + wave;
    if (gw >= batches * tilesPerBatch) return;   // wave-uniform exit

    const int n       = gw / tilesPerBatch;
    const int tile    = gw % tilesPerBatch;
    const int rowBase = tile * 32;

    const float* Xb = X + (size_t)n * R * 3;
    const v2f*   Pb = (const v2f*)panels + (size_t)n * stiles * 32;
    float*       Ob = outMin + (size_t)n * R;

    const int nl = lane & 15;
    const int hi = lane >> 4;

    // Two A operands: rows [rowBase, +16) and [rowBase+16, +32)
    v2f a0, a1;
    {
        int r0 = rowBase + nl;      if (r0 > R - 1) r0 = R - 1;
        int r1 = rowBase + 16 + nl; if (r1 > R - 1) r1 = R - 1;
        const float* p0 = Xb + (size_t)r0 * 3;
        const float* p1 = Xb + (size_t)r1 * 3;
        a0.x = hi ? p0[2] : p0[0];  a0.y = hi ? 1.0f : p0[1];
        a1.x = hi ? p1[2] : p1[0];  a1.y = hi ? 1.0f : p1[1];
    }

    v8f m0, m1;
    #pragma unroll
    for (int r = 0; r < 8; ++r) { m0[r] = 3.0e38f; m1[r] = 3.0e38f; }

    const v2f* bp = Pb + lane;
    for (int jt = 0; jt < stiles; ++jt) {
        v2f b = bp[(size_t)jt * 32];
        v8f c0 = {};
        c0 = __builtin_amdgcn_wmma_f32_16x16x4_f32(
                 false, a0, false, b, (short)0, c0, false, false);
        v8f c1 = {};
        c1 = __builtin_amdgcn_wmma_f32_16x16x4_f32(
                 false, a1, false, b, (short)0, c1, false, false);
        #pragma unroll
        for (int r = 0; r < 8; ++r) {
            m0[r] = fminf(m0[r], c0[r]);
            m1[r] = fminf(m1[r], c1[r]);
        }
    }

    // Min across 16 columns (xor-reduce inside each 16-lane half).
    #pragma unroll
    for (int r = 0; r < 8; ++r) {
        float u = m0[r], v = m1[r];
        u = fminf(u, __shfl_xor(u, 1, 16));  v = fminf(v, __shfl_xor(v, 1, 16));
        u = fminf(u, __shfl_xor(u, 2, 16));  v = fminf(v, __shfl_xor(v, 2, 16));
        u = fminf(u, __shfl_xor(u, 4, 16));  v = fminf(v, __shfl_xor(v, 4, 16));
        u = fminf(u, __shfl_xor(u, 8, 16));  v = fminf(v, __shfl_xor(v, 8, 16));
        m0[r] = u; m1[r] = v;
    }

    // Lanes 0/16 write 16 rows: tile0 row = rowBase + r + 8*hi, tile1 = +16.
    if (nl == 0) {
        #pragma unroll
        for (int r = 0; r < 8; ++r) {
            int row0 = rowBase + r + 8 * hi;
            if (row0 < R) {
                const float* xq = Xb + (size_t)row0 * 3;
                float x2 = xq[0] * xq[0] + xq[1] * xq[1] + xq[2] * xq[2];
                Ob[row0] = m0[r] + x2;
            }
            int row1 = row0 + 16;
            if (row1 < R) {
                const float* xq = Xb + (size_t)row1 * 3;
                float x2 = xq[0] * xq[0] + xq[1] * xq[1] + xq[2] * xq[2];
                Ob[row1] = m1[r] + x2;
            }
        }
    }
}

// ---------------------------------------------------------------------------
// 4) Final scalar: mean(minA) + mean(minB)
// ---------------------------------------------------------------------------
__global__ void reduce_kernel(const float* __restrict__ minA, int LA,
                              const float* __restrict__ minB, int LB,
                              float* __restrict__ out) {
    __shared__ float sA[256];
    __shared__ float sB[256];
    int t = threadIdx.x;
    float aacc = 0.0f, bacc = 0.0f;
    for (int i = t; i < LA; i += 256) aacc += minA[i];
    for (int i = t; i < LB; i += 256) bacc += minB[i];
    sA[t] = aacc; sB[t] = bacc;
    __syncthreads();
    for (int s = 128; s > 0; s >>= 1) {
        if (t < s) { sA[t] += sA[t + s]; sB[t] += sB[t + s]; }
        __syncthreads();
    }
    if (t == 0) out[0] = sA[0] / (float)LA + sB[0] / (float)LB;
}

extern "C" void kernel_launch(void* const* d_in, const int* in_sizes, int n_in,
                              void* d_out, int out_size, void* d_ws, size_t ws_size,
                              hipStream_t stream) {
    (void)in_sizes; (void)n_in; (void)out_size; (void)ws_size;

    const float* vertices = (const float*)d_in[0];   // [2,3,48,48]
    const float* pc       = (const float*)d_in[1];   // [2,8192,3]
    float*       out      = (float*)d_out;           // scalar

    const int stilesMesh = (QMESH + 15) / 16;        // 565 column tiles (mesh side)
    const int stilesPc   = (PPC + 15) / 16;          // 512 column tiles (pc side)

    char* ws = (char*)d_ws;
    size_t off = 0;
    float* mesh = (float*)(ws + off);                                 // [2,9025,3]
    off += (((size_t)NB * QMESH * 3 * sizeof(float)) + 511) & ~(size_t)511;
    float* panMesh = (float*)(ws + off);                              // [2,565,32,2]
    off += (((size_t)NB * stilesMesh * 64 * sizeof(float)) + 511) & ~(size_t)511;
    float* panPc = (float*)(ws + off);                                // [2,512,32,2]
    off += (((size_t)NB * stilesPc * 64 * sizeof(float)) + 511) & ~(size_t)511;
    float* minA = (float*)(ws + off);                                 // [2,8192]
    off += (((size_t)NB * PPC * sizeof(float)) + 511) & ~(size_t)511;
    float* minB = (float*)(ws + off);                                 // [2,9025]

    // 1) refine vertices -> mesh point list
    int totalR = NB * QMESH;
    refine_kernel<<<(totalR + 255) / 256, 256, 0, stream>>>(vertices, mesh);

    // 2) pack WMMA B panels for both directions
    int tpMesh = NB * stilesMesh * 32;
    pack_kernel<<<(tpMesh + 255) / 256, 256, 0, stream>>>(mesh, QMESH, stilesMesh, panMesh);
    int tpPc = NB * stilesPc * 32;
    pack_kernel<<<(tpPc + 255) / 256, 256, 0, stream>>>(pc, PPC, stilesPc, panPc);

    // 3) for each pc point: nearest mesh point (32 rows / wave)
    int tilesA = (PPC + 31) / 32;          // 256
    int wavesA = NB * tilesA;              // 512
    nearest_kernel<<<(wavesA + 3) / 4, 128, 0, stream>>>(pc, PPC, panMesh, stilesMesh,
                                                         minA, NB, tilesA);

    // 4) for each mesh point: nearest pc point
    int tilesB = (QMESH + 31) / 32;        // 283
    int wavesB = NB * tilesB;              // 566
    nearest_kernel<<<(wavesB + 3) / 4, 128, 0, stream>>>(mesh, QMESH, panPc, stilesPc,
                                                         minB, NB, tilesB);

    // 5) combine means
    reduce_kernel<<<1, 256, 0, stream>>>(minA, NB * PPC, minB, NB * QMESH, out);
}